// Nonlinear_Mixing_layer_torch_63831803953592
// MI455X (gfx1250) — compile-verified
//
#include <hip/hip_runtime.h>
#include <cstddef>

typedef float v2f __attribute__((ext_vector_type(2)));
typedef float v8f __attribute__((ext_vector_type(8)));

#define NPIX (512 * 512)   // H*W
#define NBAND 224          // B
#define NP 12              // P endmembers

// out[b,pix] = L + 0.5*(L*L - Q)
//   L = sum_p E[b,p]   * A[p,pix]          (linear term)
//   Q = sum_p E[b,p]^2 * A[p,pix]^2
// Identity: sum_{i<j} E_i E_j A_i A_j = (L^2 - Q)/2  (Fan-model bilinear term)
__global__ __launch_bounds__(224) void fan_mix_wmma(const float* __restrict__ E,
                                                    const float* __restrict__ A,
                                                    float* __restrict__ out)
{
    const int lane  = threadIdx.x & 31;
    const int wave  = threadIdx.x >> 5;        // 0..6  -> two 16-row M tiles each
    const int col   = lane & 15;               // N within 16-pixel tile
    const int khalf = (lane >> 4) << 1;        // 0 for lanes 0-15, 2 for lanes 16-31
    const size_t pix = (size_t)blockIdx.x * 16 + col;

    // ---- B operands: A[p, pix] in V_WMMA_F32_16X16X4 B layout, 3 k-steps ----
    // V0 holds row K=khalf, V1 holds row K=khalf+1 (per 4x16 B striping).
    v2f bL[3], bQ[3];
#pragma unroll
    for (int kk = 0; kk < 3; ++kk) {
        const int k0 = 4 * kk + khalf;
        const float b0 = A[(size_t)k0 * NPIX + pix];
        const float b1 = A[(size_t)(k0 + 1) * NPIX + pix];
        bL[kk] = (v2f){b0, b1};
        bQ[kk] = (v2f){b0 * b0, b1 * b1};      // Hadamard square for the Q GEMM
    }

    const int m0 = wave * 32;                  // this wave's 32 bands
#pragma unroll
    for (int t = 0; t < 2; ++t) {
        const int row = m0 + t * 16 + col;     // A-operand: lane%16 selects M row
        v8f accL = {};
        v8f accQ = {};
#pragma unroll
        for (int kk = 0; kk < 3; ++kk) {
            const int k0 = 4 * kk + khalf;     // A-operand: lane half selects K pair
            const float a0 = E[row * NP + k0];
            const float a1 = E[row * NP + k0 + 1];
            const v2f aL = (v2f){a0, a1};
            const v2f aQ = (v2f){a0 * a0, a1 * a1};
            // D = A x B + C, full f32 (exact reference numerics)
            accL = __builtin_amdgcn_wmma_f32_16x16x4_f32(
                false, aL, false, bL[kk], (short)0, accL, false, false);
            accQ = __builtin_amdgcn_wmma_f32_16x16x4_f32(
                false, aQ, false, bQ[kk], (short)0, accQ, false, false);
        }

        // C/D layout: VGPR r, lanes 0-15 -> M = r, lanes 16-31 -> M = r + 8
        const int mbase = m0 + t * 16 + ((lane >> 4) << 3);
#pragma unroll
        for (int r = 0; r < 8; ++r) {
            const float L = accL[r];
            const float Q = accQ[r];
            out[(size_t)(mbase + r) * NPIX + pix] = L + 0.5f * (L * L - Q);
        }
    }
}

extern "C" void kernel_launch(void* const* d_in, const int* in_sizes, int n_in,
                              void* d_out, int out_size, void* d_ws, size_t ws_size,
                              hipStream_t stream) {
    (void)in_sizes; (void)n_in; (void)d_ws; (void)ws_size; (void)out_size;
    const float* E = (const float*)d_in[0];   // (1, 224, 12) f32
    const float* A = (const float*)d_in[1];   // (1, 12, 512, 512) f32
    float* out = (float*)d_out;               // (1, 224, 512, 512) f32

    dim3 grid(NPIX / 16);   // 16384 pixel tiles
    dim3 block(224);        // 7 waves, EXEC all-ones for WMMA
    hipLaunchKernelGGL(fan_mix_wmma, grid, block, 0, stream, E, A, out);
}